// HazardRNN_17446157156386
// MI455X (gfx1250) — compile-verified
//
#include <hip/hip_runtime.h>
#include <math.h>

// Problem dims (fixed by the reference)
#define B_   128
#define T_   256
#define H_   5000
#define OUT_ 20
#define NPAD 32    // OUT padded to 2 WMMA tiles

typedef __attribute__((ext_vector_type(2))) float v2f;
typedef __attribute__((ext_vector_type(8))) float v8f;

// ---------------------------------------------------------------------------
// gfx1250 hardware tanh (transcendental unit), with safe fallback.
// ---------------------------------------------------------------------------
__device__ __forceinline__ float dev_tanh(float x) {
#if __has_builtin(__builtin_amdgcn_tanhf)
  return __builtin_amdgcn_tanhf(x);
#elif __has_builtin(__builtin_amdgcn_tanh_f32)
  return __builtin_amdgcn_tanh_f32(x);
#else
  return tanhf(x);
#endif
}

// ---------------------------------------------------------------------------
// Kernel 1: time recurrence. One lane per (b, j) element. Lane layout j-fast
// so W_in/b_in loads coalesce and the x[b,t] load is a wave-wide broadcast.
// x loaded as float4 (8 steps per 2 loads) to cut VMEM traffic; the
// tanh->fma chain is inherently serial in t.
// ---------------------------------------------------------------------------
__global__ void rnn_scan_kernel(const float* __restrict__ x,
                                const float* __restrict__ W_in,
                                const float* __restrict__ b_in,
                                float* __restrict__ h_final) {
  const int tid = blockIdx.x * blockDim.x + threadIdx.x;  // 0 .. B_*H_-1
  if (tid >= B_ * H_) return;
  const int j = tid % H_;
  const int b = tid / H_;

  const float w    = W_in[j];
  const float bias = b_in[j];
  const float4* xr = (const float4*)(x + b * T_);

  float h = 0.0f;
#pragma unroll 1
  for (int q = 0; q < T_ / 8; ++q) {
    float4 x0 = xr[2 * q + 0];
    float4 x1 = xr[2 * q + 1];
    h = dev_tanh(fmaf(x0.x, w, h + bias));
    h = dev_tanh(fmaf(x0.y, w, h + bias));
    h = dev_tanh(fmaf(x0.z, w, h + bias));
    h = dev_tanh(fmaf(x0.w, w, h + bias));
    h = dev_tanh(fmaf(x1.x, w, h + bias));
    h = dev_tanh(fmaf(x1.y, w, h + bias));
    h = dev_tanh(fmaf(x1.z, w, h + bias));
    h = dev_tanh(fmaf(x1.w, w, h + bias));
  }
  h_final[tid] = h;  // [B, H] row-major
}

// ---------------------------------------------------------------------------
// Kernel 2: logits = h_final @ W_out^T via V_WMMA_F32_16X16X4_F32 (full f32).
// One wave (32 threads) per 16x16 output tile; 8 M-tiles x 2 N-tiles.
// ISA layouts (7.12.2):
//   A 16x4 : lane m = L%16, holds A[m][kb+0..1] in v0..v1, kb = (L/16)*2
//   B 4x16 : lane n = L%16, holds B[kb+0..1][n] in v0..v1
//   D 16x16: lane n = L%16, VGPR g holds D[g + 8*(L/16)][n]
// Invalid N columns (>= OUT_) are masked by multiplying loads with 0, so
// control flow stays uniform and EXEC is all-1s around the WMMA.
// ---------------------------------------------------------------------------
__global__ void proj_wmma_kernel(const float* __restrict__ h_final,
                                 const float* __restrict__ W_out,
                                 float* __restrict__ logits /*[B_][NPAD]*/) {
  const int lane = threadIdx.x;          // 0..31
  const int mt   = blockIdx.x & 7;       // 8 M tiles
  const int nt   = blockIdx.x >> 3;      // 2 N tiles

  const int l16  = lane & 15;
  const int klo  = (lane >> 4) * 2;      // 0 or 2

  const int mrow = mt * 16 + l16;
  const int ncol = nt * 16 + l16;
  const int nc   = (ncol < OUT_) ? ncol : 0;
  const float msk = (ncol < OUT_) ? 1.0f : 0.0f;

  const float* __restrict__ arow = h_final + mrow * H_ + klo;
  const float* __restrict__ brow = W_out  + nc   * H_ + klo;

  v8f acc = {};
#pragma unroll 4
  for (int k0 = 0; k0 < H_; k0 += 4) {
    v2f a, bb;
    a.x  = arow[k0];
    a.y  = arow[k0 + 1];
    bb.x = brow[k0] * msk;
    bb.y = brow[k0 + 1] * msk;
    // 8 args: (neg_a, A, neg_b, B, c_mod, C, reuse_a, reuse_b)
    acc = __builtin_amdgcn_wmma_f32_16x16x4_f32(
        false, a, false, bb, (short)0, acc, false, false);
  }

  // Scatter D: VGPR g -> row (mt*16 + g + 8*(lane>>4)), col ncol
  const int mbase = mt * 16 + 8 * (lane >> 4);
#pragma unroll
  for (int g = 0; g < 8; ++g) {
    logits[(mbase + g) * NPAD + ncol] = acc[g];
  }
}

// ---------------------------------------------------------------------------
// Kernel 3: bias + softmax over 20 classes; one thread per batch row.
// ---------------------------------------------------------------------------
__global__ void softmax_kernel(const float* __restrict__ logits /*[B_][NPAD]*/,
                               const float* __restrict__ b_out,
                               float* __restrict__ out /*[B_][OUT_]*/) {
  const int b = blockIdx.x * blockDim.x + threadIdx.x;
  if (b >= B_) return;

  float l[OUT_];
  float m = -INFINITY;
#pragma unroll
  for (int n = 0; n < OUT_; ++n) {
    l[n] = logits[b * NPAD + n] + b_out[n];
    m = fmaxf(m, l[n]);
  }
  float s = 0.0f;
#pragma unroll
  for (int n = 0; n < OUT_; ++n) {
    l[n] = __expf(l[n] - m);
    s += l[n];
  }
  const float inv = 1.0f / s;
#pragma unroll
  for (int n = 0; n < OUT_; ++n) {
    out[b * OUT_ + n] = l[n] * inv;
  }
}

// ---------------------------------------------------------------------------
extern "C" void kernel_launch(void* const* d_in, const int* in_sizes, int n_in,
                              void* d_out, int out_size, void* d_ws, size_t ws_size,
                              hipStream_t stream) {
  const float* x     = (const float*)d_in[0];  // [128,256,1]
  const float* W_in  = (const float*)d_in[1];  // [5000,1]
  const float* b_in  = (const float*)d_in[2];  // [5000]
  const float* W_out = (const float*)d_in[3];  // [20,5000]
  const float* b_out = (const float*)d_in[4];  // [20]
  float* out = (float*)d_out;                  // [128,20]

  // Workspace carve-out: h_final [128*5000] f32, then padded logits [128*32].
  float* h_final = (float*)d_ws;
  float* logits  = h_final + (size_t)B_ * H_;

  // 1) Recurrence: 640,000 lanes, 2500 blocks of 256 (8 waves/block).
  rnn_scan_kernel<<<(B_ * H_ + 255) / 256, 256, 0, stream>>>(x, W_in, b_in, h_final);

  // 2) f32 WMMA projection: 16 single-wave blocks (8 M-tiles x 2 N-tiles).
  proj_wmma_kernel<<<16, 32, 0, stream>>>(h_final, W_out, logits);

  // 3) Bias + softmax.
  softmax_kernel<<<1, B_, 0, stream>>>(logits, b_out, out);
}